// Net_83837761618431
// MI455X (gfx1250) — compile-verified
//
#include <hip/hip_runtime.h>
#include <hip/hip_bf16.h>

// ---------------------------------------------------------------------------
// MI455X (gfx1250) GNN pipeline.
// Roofline: edge scatter/gather ~3-4 GB >> 16 GFLOP of GEMM -> HBM-bound
// (~150-200us @ 23.3 TB/s). GEMMs stay in full f32 on V_WMMA_F32_16X16X4_F32
// (exact reference numerics; matrix throughput is not the bottleneck).
// Weights are pre-transposed+zero-padded into workspace so every WMMA K-step
// is guard-free: ds_load(A from LDS) + global_load_b64(B pair) + v_wmma.
// The GAT projection stages its input tile with GLOBAL_LOAD_ASYNC_TO_LDS_B128
// (ASYNCcnt) -- the CDNA5 async global->LDS path.
// ---------------------------------------------------------------------------

typedef __attribute__((ext_vector_type(2))) float v2f;
typedef __attribute__((ext_vector_type(8))) float v8f;

#define DIN  165
#define DINP 168   // DIN padded to multiple of 4
#define DH1  82    // booster hidden (DIN//2)
#define DH1P 84    // padded
#define DH   128   // SAGE out
#define DG   64    // GAT out (DH//2)

__device__ __forceinline__ v8f wmma_k4(float a0, float a1, float b0, float b1, v8f acc) {
  v2f a; a.x = a0; a.y = a1;
  v2f b; b.x = b0; b.y = b1;
  // D = A(16x4) * B(4x16) + C, f32, wave32
  return __builtin_amdgcn_wmma_f32_16x16x4_f32(false, a, false, b, (short)0, acc, false, false);
}

// monotonic float<->uint encoding for atomic float max
__device__ __forceinline__ unsigned fenc(float f) {
  unsigned u = __float_as_uint(f);
  return (u & 0x80000000u) ? ~u : (u | 0x80000000u);
}
__device__ __forceinline__ float fdec(unsigned e) {
  unsigned u = (e & 0x80000000u) ? (e & 0x7fffffffu) : ~e;
  return __uint_as_float(u);
}

// CDNA5 async global->LDS copy, 16 bytes per lane (tracked by ASYNCcnt)
__device__ __forceinline__ void async_ld_b128(unsigned lds_off, const void* gaddr) {
  asm volatile("global_load_async_to_lds_b128 %0, %1, off"
               :: "v"(lds_off), "v"(gaddr) : "memory");
}
__device__ __forceinline__ void wait_async0() {
  asm volatile("s_wait_asynccnt 0x0" ::: "memory");
}

// ---------------- Weight pre-pass: transpose + zero-pad ----------------------
// wt[j][k] (j<NP rows, k<KP cols) = w[k*N + j] if k<K && j<N else 0
__global__ void k_padw(const float* __restrict__ w, float* __restrict__ wt,
                       int K, int N, int KP, int NP) {
  int i = blockIdx.x * blockDim.x + threadIdx.x;
  if (i >= NP * KP) return;
  int j = i / KP, k = i % KP;
  wt[i] = (k < K && j < N) ? w[(size_t)k * N + j] : 0.f;
}

// ---------------- Stage 1: FeatureBooster -----------------------------------
// xb = x * sigmoid(2 * (relu(x@w1) @ w2)); w1t:[96][DINP], w2t:[176][DH1P]
__global__ __launch_bounds__(64) void k_boost(const float* __restrict__ x,
    const float* __restrict__ w1t, const float* __restrict__ w2t,
    float* __restrict__ xb, int n)
{
  __shared__ __align__(16) float Xs[2][16][DINP];
  __shared__ __align__(16) float Hs[2][16][DH1P];
  const int wid  = threadIdx.x >> 5;
  const int lane = threadIdx.x & 31;
  const int half = lane >> 4;
  const int ln   = lane & 15;
  const int r0   = (blockIdx.x * 2 + wid) * 16;

  for (int i = lane; i < 16 * DINP; i += 32) {
    int r = i / DINP, c = i % DINP;
    float v = 0.f;
    if (c < DIN && (r0 + r) < n) v = x[(size_t)(r0 + r) * DIN + c];
    Xs[wid][r][c] = v;
  }
  __syncthreads();

  // H = relu(X @ w1): 16 x 82 (6 col tiles of 16); guard-free K loop
  for (int nt = 0; nt < 6; ++nt) {
    v8f acc = {};
    int ncol = nt * 16 + ln;
    const float* bp = w1t + (size_t)ncol * DINP;
    for (int k = 0; k < DINP; k += 4) {
      int ka = k + 2 * half;
      acc = wmma_k4(Xs[wid][ln][ka], Xs[wid][ln][ka + 1], bp[ka], bp[ka + 1], acc);
    }
#pragma unroll
    for (int i = 0; i < 8; ++i) {
      int row = i + 8 * half;
      if (ncol < DH1P) Hs[wid][row][ncol] = fmaxf(acc[i], 0.f);
    }
  }
  __syncthreads();

  // Y = H @ w2 (16 x 165, 11 col tiles); xb = x * sigmoid(2Y)
  for (int nt = 0; nt < 11; ++nt) {
    v8f acc = {};
    int ncol = nt * 16 + ln;
    const float* bp = w2t + (size_t)ncol * DH1P;
    for (int k = 0; k < DH1P; k += 4) {
      int ka = k + 2 * half;
      acc = wmma_k4(Hs[wid][ln][ka], Hs[wid][ln][ka + 1], bp[ka], bp[ka + 1], acc);
    }
#pragma unroll
    for (int i = 0; i < 8; ++i) {
      int row = i + 8 * half;
      if (ncol < DIN && (r0 + row) < n) {
        float xv = Xs[wid][row][ncol];
        float sg = 1.f / (1.f + __expf(-2.f * acc[i]));
        xb[(size_t)(r0 + row) * DIN + ncol] = xv * sg;
      }
    }
  }
}

// ---------------- Stage 2a: SAGE edge scatter (mean numerator + degree) -----
__global__ __launch_bounds__(256) void k_sage_scatter(const int* __restrict__ ei,
    const float* __restrict__ xb, float* __restrict__ agg, float* __restrict__ deg, int E)
{
  int e = blockIdx.x * 8 + (threadIdx.x >> 5);
  if (e >= E) return;
  int lane = threadIdx.x & 31;
  int s = ei[e], d = ei[E + e];
  for (int c = lane; c < DIN; c += 32)
    atomicAdd(&agg[(size_t)d * DIN + c], xb[(size_t)s * DIN + c]);
  if (lane == 0) atomicAdd(&deg[d], 1.f);
}

// ---------------- Stage 2b: SAGE: x2 = relu((agg/deg)@wl + bl + xb@wr) ------
// wlt/wrt: [DH][DINP]
__global__ __launch_bounds__(64) void k_sage(const float* __restrict__ xb,
    const float* __restrict__ agg, const float* __restrict__ deg,
    const float* __restrict__ wlt, const float* __restrict__ bl,
    const float* __restrict__ wrt, float* __restrict__ x2, int n)
{
  __shared__ __align__(16) float As[2][16][DINP];
  __shared__ __align__(16) float Xs[2][16][DINP];
  const int wid  = threadIdx.x >> 5;
  const int lane = threadIdx.x & 31;
  const int half = lane >> 4;
  const int ln   = lane & 15;
  const int r0   = (blockIdx.x * 2 + wid) * 16;

  for (int i = lane; i < 16 * DINP; i += 32) {
    int r = i / DINP, c = i % DINP;
    float av = 0.f, xv = 0.f;
    if (c < DIN && (r0 + r) < n) {
      float rd = 1.f / fmaxf(deg[r0 + r], 1.f);
      av = agg[(size_t)(r0 + r) * DIN + c] * rd;
      xv = xb[(size_t)(r0 + r) * DIN + c];
    }
    As[wid][r][c] = av;
    Xs[wid][r][c] = xv;
  }
  __syncthreads();

  for (int nt = 0; nt < 8; ++nt) {  // DH=128 -> 8 col tiles
    v8f acc = {};
    int ncol = nt * 16 + ln;
    const float* bpl = wlt + (size_t)ncol * DINP;
    const float* bpr = wrt + (size_t)ncol * DINP;
    for (int k = 0; k < DINP; k += 4) {
      int ka = k + 2 * half;
      acc = wmma_k4(As[wid][ln][ka], As[wid][ln][ka + 1], bpl[ka], bpl[ka + 1], acc);
    }
    for (int k = 0; k < DINP; k += 4) {
      int ka = k + 2 * half;
      acc = wmma_k4(Xs[wid][ln][ka], Xs[wid][ln][ka + 1], bpr[ka], bpr[ka + 1], acc);
    }
    float bv = bl[ncol];
#pragma unroll
    for (int i = 0; i < 8; ++i) {
      int row = i + 8 * half;
      if ((r0 + row) < n)
        x2[(size_t)(r0 + row) * DH + ncol] = fmaxf(acc[i] + bv, 0.f);
    }
  }
}

// ---------------- Stage 3a: GAT projection h = x2 @ gat_w; a_s/a_d dots -----
// gwt: [DG][DH]. Tile staged via async global->LDS B128 copies.
__global__ __launch_bounds__(64) void k_gatprep(const float* __restrict__ x2,
    const float* __restrict__ gwt, const float* __restrict__ asrc,
    const float* __restrict__ adst, float* __restrict__ h,
    float* __restrict__ a_s, float* __restrict__ a_d, int n)
{
  __shared__ __align__(16) float Xs[2][16][DH];
  __shared__ __align__(16) float Hsh[2][16][DG];
  const int wid  = threadIdx.x >> 5;
  const int lane = threadIdx.x & 31;
  const int half = lane >> 4;
  const int ln   = lane & 15;
  const int r0   = (blockIdx.x * 2 + wid) * 16;

  if (r0 + 15 < n) {
    // full tile: 16 rows x 512B, lane covers cols [4*lane, 4*lane+3] per row
#pragma unroll
    for (int j = 0; j < 16; ++j) {
      unsigned dst = (unsigned)(uintptr_t)&Xs[wid][j][lane * 4];
      async_ld_b128(dst, &x2[(size_t)(r0 + j) * DH + lane * 4]);
    }
  } else {
    for (int i = lane; i < 16 * DH; i += 32) {
      int r = i / DH, c = i % DH;
      Xs[wid][r][c] = ((r0 + r) < n) ? x2[(size_t)(r0 + r) * DH + c] : 0.f;
    }
  }
  wait_async0();
  __syncthreads();

  for (int nt = 0; nt < 4; ++nt) {  // DG=64 -> 4 col tiles
    v8f acc = {};
    int ncol = nt * 16 + ln;
    const float* bp = gwt + (size_t)ncol * DH;
    for (int k = 0; k < DH; k += 4) {
      int ka = k + 2 * half;
      acc = wmma_k4(Xs[wid][ln][ka], Xs[wid][ln][ka + 1], bp[ka], bp[ka + 1], acc);
    }
#pragma unroll
    for (int i = 0; i < 8; ++i) {
      int row = i + 8 * half;
      Hsh[wid][row][ncol] = acc[i];
      if ((r0 + row) < n) h[(size_t)(r0 + row) * DG + ncol] = acc[i];
    }
  }
  __syncthreads();

  if (lane < 16) {
    int r = lane;
    float sa = 0.f, sd = 0.f;
    for (int c = 0; c < DG; ++c) {
      float v = Hsh[wid][r][c];
      sa += v * asrc[c];
      sd += v * adst[c];
    }
    if ((r0 + r) < n) { a_s[r0 + r] = sa; a_d[r0 + r] = sd; }
  }
}

// ---------------- Stage 3b: segment softmax init / max / sum / scatter ------
__global__ void k_init(unsigned* __restrict__ mxe, float* __restrict__ ssum, int n) {
  int i = blockIdx.x * blockDim.x + threadIdx.x;
  if (i < n) { mxe[i] = fenc(-3.0e38f); ssum[i] = 0.f; }
}

__global__ void k_edge_max(const int* __restrict__ ei, const float* __restrict__ a_s,
    const float* __restrict__ a_d, unsigned* __restrict__ mxe, int E, int EA)
{
  int e = blockIdx.x * blockDim.x + threadIdx.x;
  if (e >= EA) return;
  int s, d;
  if (e < E) { s = ei[e]; d = ei[E + e]; } else { s = d = e - E; }  // self loops
  float l = a_s[s] + a_d[d];
  l = (l > 0.f) ? l : 0.2f * l;  // leaky_relu(0.2)
  atomicMax(&mxe[d], fenc(l));
}

__global__ void k_edge_sum(const int* __restrict__ ei, const float* __restrict__ a_s,
    const float* __restrict__ a_d, const unsigned* __restrict__ mxe,
    float* __restrict__ ssum, int E, int EA)
{
  int e = blockIdx.x * blockDim.x + threadIdx.x;
  if (e >= EA) return;
  int s, d;
  if (e < E) { s = ei[e]; d = ei[E + e]; } else { s = d = e - E; }
  float l = a_s[s] + a_d[d];
  l = (l > 0.f) ? l : 0.2f * l;
  atomicAdd(&ssum[d], __expf(l - fdec(mxe[d])));
}

__global__ __launch_bounds__(256) void k_edge_scatter(const int* __restrict__ ei,
    const float* __restrict__ a_s, const float* __restrict__ a_d,
    const unsigned* __restrict__ mxe, const float* __restrict__ ssum,
    const float* __restrict__ h, float* __restrict__ o, int E, int EA)
{
  int e = blockIdx.x * 8 + (threadIdx.x >> 5);
  if (e >= EA) return;
  int lane = threadIdx.x & 31;
  int s, d;
  if (e < E) { s = ei[e]; d = ei[E + e]; } else { s = d = e - E; }
  __builtin_prefetch(&h[(size_t)s * DG + lane], 0, 1);  // global_prefetch_b8
  float l = a_s[s] + a_d[d];
  l = (l > 0.f) ? l : 0.2f * l;
  float alpha = __expf(l - fdec(mxe[d])) / (ssum[d] + 1e-16f);
  for (int c = lane; c < DG; c += 32)
    atomicAdd(&o[(size_t)d * DG + c], h[(size_t)s * DG + c] * alpha);
}

// ---------------- Stage 4: relu + Cheb(K=1) linear + sigmoid ----------------
__global__ __launch_bounds__(256) void k_final(const float* __restrict__ o,
    const float* __restrict__ gb, const float* __restrict__ cw,
    const float* __restrict__ cb, float* __restrict__ out, int n)
{
  int idx = blockIdx.x * 8 + (threadIdx.x >> 5);
  int lane = threadIdx.x & 31;
  if (idx >= n) return;
  float v = 0.f;
  for (int c = lane; c < DG; c += 32)
    v += fmaxf(o[(size_t)idx * DG + c] + gb[c], 0.f) * cw[c];
  for (int off = 16; off > 0; off >>= 1) v += __shfl_xor(v, off, 32);
  if (lane == 0) out[idx] = 1.f / (1.f + __expf(-(v + cb[0])));
}

// ---------------------------------------------------------------------------
extern "C" void kernel_launch(void* const* d_in, const int* in_sizes, int n_in,
                              void* d_out, int out_size, void* d_ws, size_t ws_size,
                              hipStream_t stream) {
  (void)n_in; (void)out_size; (void)ws_size;
  const float* x       = (const float*)d_in[0];
  const int*   ei      = (const int*)d_in[1];
  const float* fb_w1   = (const float*)d_in[2];
  const float* fb_w2   = (const float*)d_in[3];
  const float* sage_wl = (const float*)d_in[4];
  const float* sage_bl = (const float*)d_in[5];
  const float* sage_wr = (const float*)d_in[6];
  const float* gat_w   = (const float*)d_in[7];
  const float* gat_as  = (const float*)d_in[8];
  const float* gat_ad  = (const float*)d_in[9];
  const float* gat_b   = (const float*)d_in[10];
  const float* cheb_w  = (const float*)d_in[11];
  const float* cheb_b  = (const float*)d_in[12];

  const int n  = in_sizes[0] / DIN;
  const int E  = in_sizes[1] / 2;
  const int EA = E + n;  // + self loops for GAT

  // workspace layout, 16B-aligned slabs (peak ~184 MB + ~330 KB weights)
  float* ws = (float*)d_ws;
  size_t off = 0;
  auto alloc = [&](size_t cnt) { float* p = ws + off; off += (cnt + 3) & ~(size_t)3; return p; };
  float* xb  = alloc((size_t)n * DIN);      // boosted features
  float* agg = alloc((size_t)n * DIN);      // SAGE sum
  float* deg = alloc((size_t)n);
  float* x2  = alloc((size_t)n * DH);       // SAGE out (row stride 512B)
  float* w1t = alloc((size_t)96  * DINP);   // transposed+padded weights
  float* w2t = alloc((size_t)176 * DH1P);
  float* wlt = alloc((size_t)DH  * DINP);
  float* wrt = alloc((size_t)DH  * DINP);
  float* gwt = alloc((size_t)DG  * DH);
  // reuse of dead slabs:
  float* h   = xb;                          // [n, DG]  (xb dead after k_sage)
  float* a_s = agg;                         // [n]      (agg dead after k_sage)
  float* a_d = agg + (size_t)n;
  unsigned* mxe = (unsigned*)(agg + 2 * (size_t)n);
  float* ssum   = agg + 3 * (size_t)n;
  float* o      = x2;                       // [n, DG]  (x2 dead after k_gatprep)
  float* out    = (float*)d_out;

  const int tiles = (n + 15) / 16;
  const int gblk  = (tiles + 1) / 2;  // 2 waves (2 row-tiles) per block

  // weight transpose/pad pre-pass (tiny; L2-resident afterwards)
  k_padw<<<(96 * DINP + 255) / 256, 256, 0, stream>>>(fb_w1, w1t, DIN, DH1, DINP, 96);
  k_padw<<<(176 * DH1P + 255) / 256, 256, 0, stream>>>(fb_w2, w2t, DH1, DIN, DH1P, 176);
  k_padw<<<(DH * DINP + 255) / 256, 256, 0, stream>>>(sage_wl, wlt, DIN, DH, DINP, DH);
  k_padw<<<(DH * DINP + 255) / 256, 256, 0, stream>>>(sage_wr, wrt, DIN, DH, DINP, DH);
  k_padw<<<(DG * DH + 255) / 256, 256, 0, stream>>>(gat_w, gwt, DH, DG, DH, DG);

  k_boost<<<gblk, 64, 0, stream>>>(x, w1t, w2t, xb, n);
  hipMemsetAsync(agg, 0, (size_t)n * DIN * sizeof(float), stream);
  hipMemsetAsync(deg, 0, (size_t)n * sizeof(float), stream);
  k_sage_scatter<<<(E + 7) / 8, 256, 0, stream>>>(ei, xb, agg, deg, E);
  k_sage<<<gblk, 64, 0, stream>>>(xb, agg, deg, wlt, sage_bl, wrt, x2, n);
  k_gatprep<<<gblk, 64, 0, stream>>>(x2, gwt, gat_as, gat_ad, h, a_s, a_d, n);
  k_init<<<(n + 255) / 256, 256, 0, stream>>>(mxe, ssum, n);
  hipMemsetAsync(o, 0, (size_t)n * DG * sizeof(float), stream);
  k_edge_max<<<(EA + 255) / 256, 256, 0, stream>>>(ei, a_s, a_d, mxe, E, EA);
  k_edge_sum<<<(EA + 255) / 256, 256, 0, stream>>>(ei, a_s, a_d, mxe, ssum, E, EA);
  k_edge_scatter<<<(EA + 7) / 8, 256, 0, stream>>>(ei, a_s, a_d, mxe, ssum, h, o, E, EA);
  k_final<<<(n + 7) / 8, 256, 0, stream>>>(o, gat_b, cheb_w, cheb_b, out, n);
}